// BallQueryLayer_57913339019923
// MI455X (gfx1250) — compile-verified
//
#include <hip/hip_runtime.h>
#include <hip/hip_bf16.h>

// ---------------------------------------------------------------------------
// Ball query (radius search, first-K in index order) for MI455X / gfx1250.
//
// Stage 1: per-wave 16-row tile; two V_WMMA_F32_16X16X4_F32 per 16x16 tile
//          compute d^2 directly via the rank-5 distance embedding; ballots
//          compress "within radius" to 1 bit/pair -> 8 MB mask array in d_ws.
// Stage 2: one thread per query row scans its 256 mask words in index order,
//          extracts the first K=32 neighbor indices, writes mapping / counts /
//          gathered neighbor coordinates (zero-padded).
// ---------------------------------------------------------------------------

#define N1 8192
#define N2 8192
#define KNN 32
#define RADIUS 0.1f

typedef __attribute__((ext_vector_type(2))) float v2f;
typedef __attribute__((ext_vector_type(8))) float v8f;

// Pick this lane's (row = lane&15) 16-bit mask out of 8 wave-uniform ballots.
// Ballot bit layout (C-matrix layout): ballot of c[v] has row v in bits[15:0]
// and row v+8 in bits[31:16].
__device__ __forceinline__ unsigned pick_row_mask(int lane,
                                                  unsigned m0, unsigned m1,
                                                  unsigned m2, unsigned m3,
                                                  unsigned m4, unsigned m5,
                                                  unsigned m6, unsigned m7) {
    int r = lane & 7;
    unsigned t01 = (r & 1) ? m1 : m0;
    unsigned t23 = (r & 1) ? m3 : m2;
    unsigned t45 = (r & 1) ? m5 : m4;
    unsigned t67 = (r & 1) ? m7 : m6;
    unsigned t03 = (r & 2) ? t23 : t01;
    unsigned t47 = (r & 2) ? t67 : t45;
    unsigned tt  = (r & 4) ? t47 : t03;
    return ((lane & 8) ? (tt >> 16) : tt) & 0xFFFFu;
}

// Compute one 16x16 tile of d^2 via two K=4 f32 WMMAs (rank-5 embedding).
__device__ __forceinline__ v8f d2_tile(v2f a, v2f a2, const float* __restrict__ p2,
                                       int tile, int lane, bool hi) {
    int j = tile * 16 + (lane & 15);
    float x2 = p2[3 * j + 0];
    float y2 = p2[3 * j + 1];
    float z2 = p2[3 * j + 2];
    float s2 = x2 * x2 + y2 * y2 + z2 * z2;
    // B (4x16), VGPR0: lanes0-15 K=0, lanes16-31 K=2; VGPR1: K=1 / K=3.
    v2f b, b2;
    b.x  = hi ? (-2.0f * z2) : (-2.0f * x2);
    b.y  = hi ? 1.0f         : (-2.0f * y2);
    b2.x = hi ? 0.0f : s2;
    b2.y = 0.0f;
    v8f c = {};
    c = __builtin_amdgcn_wmma_f32_16x16x4_f32(false, a,  false, b,  (short)0, c, false, false);
    c = __builtin_amdgcn_wmma_f32_16x16x4_f32(false, a2, false, b2, (short)0, c, false, false);
    return c;
}

__global__ void __launch_bounds__(128)
ball_query_stage1(const float* __restrict__ p1, const float* __restrict__ p2,
                  unsigned* __restrict__ masks) {
    const float R2 = RADIUS * RADIUS;
    const int lane = threadIdx.x & 31;
    const int wave = blockIdx.x * (blockDim.x >> 5) + (threadIdx.x >> 5);
    const int row_base = wave * 16;          // 16 query rows per wave
    const bool hi = lane >= 16;

    // A operand (16x4 f32): lanes0-15 VGPR{0,1}=K{0,1}, lanes16-31 = K{2,3}.
    // Row m: [x, y, z, |p|^2]; second matmul row: [1, 0, 0, 0].
    int m = row_base + (lane & 15);
    float x1 = p1[3 * m + 0];
    float y1 = p1[3 * m + 1];
    float z1 = p1[3 * m + 2];
    float s1 = x1 * x1 + y1 * y1 + z1 * z1;
    v2f a, a2;
    a.x  = hi ? z1 : x1;
    a.y  = hi ? s1 : y1;
    a2.x = hi ? 0.0f : 1.0f;
    a2.y = 0.0f;

    for (int t32 = 0; t32 < N2 / 32; ++t32) {
        // Prefetch next iteration's B-operand cache lines (global_prefetch_b8).
        int jn = (t32 + 1) * 32 + (lane & 15);
        if (jn < N2) __builtin_prefetch(&p2[3 * jn], 0, 1);

        v8f cA = d2_tile(a, a2, p2, 2 * t32 + 0, lane, hi);
        v8f cB = d2_tile(a, a2, p2, 2 * t32 + 1, lane, hi);

        unsigned a0 = __builtin_amdgcn_ballot_w32(cA[0] <= R2);
        unsigned a1 = __builtin_amdgcn_ballot_w32(cA[1] <= R2);
        unsigned a2b = __builtin_amdgcn_ballot_w32(cA[2] <= R2);
        unsigned a3 = __builtin_amdgcn_ballot_w32(cA[3] <= R2);
        unsigned a4 = __builtin_amdgcn_ballot_w32(cA[4] <= R2);
        unsigned a5 = __builtin_amdgcn_ballot_w32(cA[5] <= R2);
        unsigned a6 = __builtin_amdgcn_ballot_w32(cA[6] <= R2);
        unsigned a7 = __builtin_amdgcn_ballot_w32(cA[7] <= R2);

        unsigned b0 = __builtin_amdgcn_ballot_w32(cB[0] <= R2);
        unsigned b1 = __builtin_amdgcn_ballot_w32(cB[1] <= R2);
        unsigned b2 = __builtin_amdgcn_ballot_w32(cB[2] <= R2);
        unsigned b3 = __builtin_amdgcn_ballot_w32(cB[3] <= R2);
        unsigned b4 = __builtin_amdgcn_ballot_w32(cB[4] <= R2);
        unsigned b5 = __builtin_amdgcn_ballot_w32(cB[5] <= R2);
        unsigned b6 = __builtin_amdgcn_ballot_w32(cB[6] <= R2);
        unsigned b7 = __builtin_amdgcn_ballot_w32(cB[7] <= R2);

        unsigned lo = pick_row_mask(lane, a0, a1, a2b, a3, a4, a5, a6, a7);
        unsigned hiw = pick_row_mask(lane, b0, b1, b2, b3, b4, b5, b6, b7);
        unsigned word = lo | (hiw << 16);

        // [t32][row] layout: 16 consecutive dwords per wave -> coalesced,
        // and consecutive query rows are consecutive for stage 2.
        if (lane < 16) masks[t32 * N1 + (row_base + lane)] = word;
    }
}

__global__ void __launch_bounds__(256)
ball_query_stage2(const unsigned* __restrict__ masks, const float* __restrict__ p2,
                  float* __restrict__ map_out, float* __restrict__ cnt_out,
                  float* __restrict__ pts_out) {
    int i = blockIdx.x * blockDim.x + threadIdx.x;
    if (i >= N1) return;

    int cnt = 0;
    for (int t = 0; t < N2 / 32 && cnt < KNN; ++t) {
        unsigned w = masks[t * N1 + i];
        while (w != 0u && cnt < KNN) {
            int b = __builtin_ctz(w);
            w &= w - 1u;
            int j = t * 32 + b;
            map_out[i * KNN + cnt] = (float)j;
            pts_out[(i * KNN + cnt) * 3 + 0] = p2[3 * j + 0];
            pts_out[(i * KNN + cnt) * 3 + 1] = p2[3 * j + 1];
            pts_out[(i * KNN + cnt) * 3 + 2] = p2[3 * j + 2];
            ++cnt;
        }
    }
    cnt_out[i] = (float)cnt;
    for (int c = cnt; c < KNN; ++c) {
        map_out[i * KNN + c] = 0.0f;
        pts_out[(i * KNN + c) * 3 + 0] = 0.0f;
        pts_out[(i * KNN + c) * 3 + 1] = 0.0f;
        pts_out[(i * KNN + c) * 3 + 2] = 0.0f;
    }
}

extern "C" void kernel_launch(void* const* d_in, const int* in_sizes, int n_in,
                              void* d_out, int out_size, void* d_ws, size_t ws_size,
                              hipStream_t stream) {
    const float* p1 = (const float*)d_in[0];   // [1, N1, 3] f32
    const float* p2 = (const float*)d_in[1];   // [1, N2, 3] f32
    // d_in[2], d_in[3] = lengths (unused, full clouds)

    unsigned* masks = (unsigned*)d_ws;         // N1 * (N2/32) uint32 = 8 MB

    float* out     = (float*)d_out;
    float* map_out = out;                      // N1*KNN
    float* cnt_out = out + (size_t)N1 * KNN;   // N1
    float* pts_out = cnt_out + N1;             // N1*KNN*3

    // Stage 1: 512 waves (16 query rows each), 4 waves per block.
    ball_query_stage1<<<(N1 / 16) / 4, 128, 0, stream>>>(p1, p2, masks);
    // Stage 2: one thread per query row.
    ball_query_stage2<<<N1 / 256, 256, 0, stream>>>(masks, p2, map_out, cnt_out, pts_out);
}